// Polarisation_12120397709775
// MI455X (gfx1250) — compile-verified
//
#include <hip/hip_runtime.h>

#define BOHRF   0.52917721067f
#define DAMP_M  0.39f
#define DAMP_F  0.7f
#define N_ITER  7

typedef __attribute__((ext_vector_type(2))) float v2f;
typedef __attribute__((ext_vector_type(8))) float v8f;

// ---------------------------------------------------------------- zero init
__global__ void zero_kernel(float* __restrict__ p, int n) {
    int i = blockIdx.x * blockDim.x + threadIdx.x;
    if (i < n) p[i] = 0.0f;
}

// ---------------------------------------------------------------- edge setup
// Per edge: c1 = l3m/r^3, c2 = 3*l5m/r^5, v_ij (bohr). Scatter E0 via atomics.
__global__ void edge_setup(const int* __restrict__ src, const int* __restrict__ dst,
                           const float* __restrict__ dist, const float* __restrict__ vec,
                           const float* __restrict__ pol, const float* __restrict__ q,
                           float* __restrict__ E0, float4* __restrict__ c4,
                           float* __restrict__ vz, int E) {
    int e = blockIdx.x * blockDim.x + threadIdx.x;
    if (e >= E) return;
    const float invB  = 1.0f / BOHRF;
    const float invB3 = invB * invB * invB;
    int s = src[e], d = dst[e];
    float rij = dist[e] * invB;
    float vx = vec[e*3+0] * invB;
    float vy = vec[e*3+1] * invB;
    float vzv = vec[e*3+2] * invB;
    float ps = pol[s] * invB3, pd = pol[d] * invB3;
    float rij2 = rij * rij;
    float rij3 = rij2 * rij;
    float u = rij3 * rsqrtf(ps * pd);
    float em  = expf(-DAMP_M * u);
    float l3m = 1.0f - em;
    float l5m = 1.0f - (1.0f + DAMP_M * u) * em;
    float l3f = 1.0f - expf(-DAMP_F * u);
    float invr3 = 1.0f / rij3;
    float invr5 = invr3 / rij2;
    float c1 = l3m * invr3;
    float c2 = 3.0f * l5m * invr5;
    c4[e] = make_float4(c1, c2, vx, vy);
    vz[e] = vzv;
    float sc = l3f * q[d] * invr3;
    atomicAdd(&E0[s*4+0], sc * vx);
    atomicAdd(&E0[s*4+1], sc * vy);
    atomicAdd(&E0[s*4+2], sc * vzv);
}

// ---------------------------------------------------------------- mu0 = alpha*E0
__global__ void node_init_mu(const float* __restrict__ E0, const float* __restrict__ pol,
                             float* __restrict__ mu, int N) {
    int i = blockIdx.x * blockDim.x + threadIdx.x;
    if (i >= N) return;
    const float invB = 1.0f / BOHRF;
    float pb = pol[i] * (invB * invB * invB);
    mu[i*4+0] = pb * E0[i*4+0];
    mu[i*4+1] = pb * E0[i*4+1];
    mu[i*4+2] = pb * E0[i*4+2];
    mu[i*4+3] = 0.0f;
}

// ---------------------------------------------------------------- SpMV edge kernel
// Wave-private double-buffered async staging of edge coefficients into LDS
// (global_load_async_to_lds_*), mu gathered as one b128 load, scatter via
// global_atomic_add_f32.
__global__ void __launch_bounds__(256)
edge_matvec(const int* __restrict__ src, const int* __restrict__ dst,
            const float4* __restrict__ c4, const float* __restrict__ vz,
            const float4* __restrict__ mu, float* __restrict__ field,
            int numTiles, int E) {
    __shared__ float4 sc4[8][2][32];
    __shared__ float  scz[8][2][32];
    const int lane = threadIdx.x & 31;
    const int wave = threadIdx.x >> 5;
    const int wpb  = blockDim.x >> 5;
    const int gw   = blockIdx.x * wpb + wave;
    const int stride = gridDim.x * wpb;

    auto stage = [&](int tile, int p) {
        int ee = tile * 32 + lane;
        int e  = ee < E ? ee : E - 1;
        unsigned lds4 = (unsigned)(size_t)&sc4[wave][p][lane];
        unsigned ldsz = (unsigned)(size_t)&scz[wave][p][lane];
        unsigned long long a4 = (unsigned long long)(size_t)&c4[e];
        unsigned long long az = (unsigned long long)(size_t)&vz[e];
        asm volatile("global_load_async_to_lds_b128 %0, %1, off"
                     :: "v"(lds4), "v"(a4) : "memory");
        asm volatile("global_load_async_to_lds_b32 %0, %1, off"
                     :: "v"(ldsz), "v"(az) : "memory");
    };

    int t = gw;
    int p = 0;
    if (t < numTiles) stage(t, 0);
    for (; t < numTiles; t += stride, p ^= 1) {
        int nt = t + stride;
        if (nt < numTiles) stage(nt, p ^ 1);
        // current tile's 2 async loads done (in-order completion); next tile's
        // 2 may still be in flight.
        asm volatile("s_wait_asynccnt 2" ::: "memory");
        float4 cc  = sc4[wave][p][lane];
        float  vzl = scz[wave][p][lane];
        int ee = t * 32 + lane;
        int e  = ee < E ? ee : E - 1;
        int d  = dst[e];
        int s  = src[e];
        float4 md = mu[d];
        float dot = cc.z * md.x + cc.w * md.y + vzl * md.z;
        float k   = cc.y * dot;                    // c2 * dot
        float tx = cc.x * md.x - k * cc.z;
        float ty = cc.x * md.y - k * cc.w;
        float tz = cc.x * md.z - k * vzl;
        if (ee < E) {
            atomicAdd(&field[s*4+0], tx);
            atomicAdd(&field[s*4+1], ty);
            atomicAdd(&field[s*4+2], tz);
        }
    }
    asm volatile("s_wait_asynccnt 0" ::: "memory");
}

// ---------------------------------------------------------------- node update
__global__ void node_update(const float* __restrict__ E0, const float* __restrict__ pol,
                            float* __restrict__ field, float* __restrict__ mu, int N) {
    int i = blockIdx.x * blockDim.x + threadIdx.x;
    if (i >= N) return;
    const float invB = 1.0f / BOHRF;
    float pb = pol[i] * (invB * invB * invB);
    #pragma unroll
    for (int k = 0; k < 3; ++k) {
        mu[i*4+k] = pb * (E0[i*4+k] - field[i*4+k]);
        field[i*4+k] = 0.0f;           // reset accumulator for next iteration
    }
    mu[i*4+3] = 0.0f;
}

// ---------------------------------------------------------------- energy (WMMA)
// 16 per-node dot products mu_i . E0_i as the diagonal of one 16x16x4 f32 WMMA.
// A (16x4): row M = node, K = component. Lanes 0-15 carry K=0,1; lanes 16-31 K=2,3.
// B (4x16): col N = node, same K split. D[e][e] = sum_k mu[e][k]*E0[e][k].
__global__ void __launch_bounds__(256)
energy_wmma(const float4* __restrict__ mu, const float4* __restrict__ E0,
            float* __restrict__ out, int N) {
    __shared__ float dt[8][256];
    const int lane = threadIdx.x & 31;
    const int wave = threadIdx.x >> 5;
    const int gw   = blockIdx.x * (blockDim.x >> 5) + wave;
    const int base = gw * 16;
    const int col  = lane & 15;
    int node = base + col;
    int nc = node < N ? node : N - 1;   // clamp loads; keep EXEC full for WMMA
    float4 m = mu[nc];
    float4 f = E0[nc];
    bool lo = lane < 16;
    v2f a, b;
    a[0] = lo ? m.x : m.z;   a[1] = lo ? m.y : 0.0f;
    b[0] = lo ? f.x : f.z;   b[1] = lo ? f.y : 0.0f;
    v8f c = {};
    c = __builtin_amdgcn_wmma_f32_16x16x4_f32(
            /*neg_a=*/false, a, /*neg_b=*/false, b,
            /*c_mod=*/(short)0, c, /*reuse_a=*/false, /*reuse_b=*/false);
    const int mrow = lo ? 0 : 8;
    #pragma unroll
    for (int r = 0; r < 8; ++r)
        dt[wave][(mrow + r) * 16 + col] = c[r];
    __syncthreads();
    if (lo && node < N)
        out[node] = -0.5f * dt[wave][col * 17];   // diagonal element (col,col)
}

// ---------------------------------------------------------------- launcher
extern "C" void kernel_launch(void* const* d_in, const int* in_sizes, int n_in,
                              void* d_out, int out_size, void* d_ws, size_t ws_size,
                              hipStream_t stream) {
    const int*   esrc = (const int*)d_in[1];
    const int*   edst = (const int*)d_in[2];
    const float* dist = (const float*)d_in[3];
    const float* vec  = (const float*)d_in[4];
    const float* pol  = (const float*)d_in[5];
    const float* q    = (const float*)d_in[6];
    const int E = in_sizes[3];
    const int N = in_sizes[5];

    float* ws    = (float*)d_ws;
    float* E0    = ws;                                   // 4N
    float* field = ws + 4 * (size_t)N;                   // 4N
    float* muv   = ws + 8 * (size_t)N;                   // 4N
    float4* c4   = (float4*)(ws + 12 * (size_t)N);       // 4E floats (16B aligned)
    float* vz    = ws + 12 * (size_t)N + 4 * (size_t)E;  // E

    // zero E0 + field (must happen every call: atomics accumulate into them)
    {
        int n = 8 * N;
        zero_kernel<<<(n + 255) / 256, 256, 0, stream>>>(E0, n);
    }
    edge_setup<<<(E + 255) / 256, 256, 0, stream>>>(esrc, edst, dist, vec, pol, q,
                                                    E0, c4, vz, E);
    node_init_mu<<<(N + 255) / 256, 256, 0, stream>>>(E0, pol, muv, N);

    const int numTiles = (E + 31) / 32;
    for (int it = 0; it < N_ITER; ++it) {
        edge_matvec<<<1024, 256, 0, stream>>>(esrc, edst, c4, vz,
                                              (const float4*)muv, field, numTiles, E);
        node_update<<<(N + 255) / 256, 256, 0, stream>>>(E0, pol, field, muv, N);
    }

    const int waves  = (N + 15) / 16;
    const int blocks = (waves + 7) / 8;
    energy_wmma<<<blocks, 256, 0, stream>>>((const float4*)muv, (const float4*)E0,
                                            (float*)d_out, N);
}